// Att_86595130622352
// MI455X (gfx1250) — compile-verified
//
#include <hip/hip_runtime.h>

typedef float v2f __attribute__((ext_vector_type(2)));
typedef float v8f __attribute__((ext_vector_type(8)));

// CDNA5 native V_TANH_F32 when clang exposes it.
#if defined(__has_builtin)
# if __has_builtin(__builtin_amdgcn_tanhf)
#  define FAST_TANH(x) __builtin_amdgcn_tanhf(x)
# elif __has_builtin(__builtin_amdgcn_tanh_f32)
#  define FAST_TANH(x) __builtin_amdgcn_tanh_f32(x)
# endif
#endif
#ifndef FAST_TANH
# define FAST_TANH(x) tanhf(x)
#endif

#define WMMA_F32(A, B, C) \
    __builtin_amdgcn_wmma_f32_16x16x4_f32(false, (A), false, (B), (short)0, (C), false, false)

// One recurrence step (transposed GEMM, duplicated-Wout trick):
//   c = cpre + Wout_dup x o1n_prev^T  (2 chained WMMAs, serial part)
//   cpre = bias + Wout_dup x ctx(TN)^T (2 WMMAs, independent -> overlaps tanh)
//   act/y from c. Every lane holds o1n[col][v], col = lane%16.
#define STEP(YVAR, TN, DO_PRE)                                                             \
    do {                                                                                   \
        v2f fb0_ = { lo ? a0 : a2, lo ? a1 : a3 };                                         \
        v2f fb1_ = { lo ? a4 : a6, lo ? a5 : a7 };                                         \
        v8f c1_ = WMMA_F32(A2, fb0_, cpre);                                                \
        v8f c2_ = WMMA_F32(A3, fb1_, c1_);                                                 \
        if (DO_PRE) {                                                                      \
            unsigned tn_ = (TN);                                                           \
            v2f nb0_ = { in[uw0 + tn_],     in[uw1 + tn_] };                               \
            v2f nb1_ = { in[uw0 + tn_ + 4], in[uw1 + tn_ + 4] };                           \
            v8f p1_ = WMMA_F32(A0, nb0_, cb);                                              \
            cpre    = WMMA_F32(A1, nb1_, p1_);                                             \
        }                                                                                  \
        a0 = fmaxf(FAST_TANH(c2_[0]), 0.f);                                                \
        a1 = fmaxf(FAST_TANH(c2_[1]), 0.f);                                                \
        a2 = fmaxf(FAST_TANH(c2_[2]), 0.f);                                                \
        a3 = fmaxf(FAST_TANH(c2_[3]), 0.f);                                                \
        a4 = fmaxf(FAST_TANH(c2_[4]), 0.f);                                                \
        a5 = fmaxf(FAST_TANH(c2_[5]), 0.f);                                                \
        a6 = fmaxf(FAST_TANH(c2_[6]), 0.f);                                                \
        a7 = fmaxf(FAST_TANH(c2_[7]), 0.f);                                                \
        YVAR = a0 * w10 + a1 * w11 + a2 * w12 + a3 * w13                                   \
             + a4 * w14 + a5 * w15 + a6 * w16 + a7 * w17 + b1v;                            \
    } while (0)

__global__ __launch_bounds__(32)
void att_rnn_wmma(const float* __restrict__ in,
                  const float* __restrict__ Wout,   // [8][16] row-major
                  const float* __restrict__ bout,   // [8]
                  const float* __restrict__ w1,     // [8]
                  const float* __restrict__ b1p,    // [1]
                  float* __restrict__ out,          // [N][T]
                  int T) {
    const int L = 8;
    const int S = T - L;                     // 991 recurrence steps

    const int      lane = threadIdx.x & 31;
    const bool     lo   = lane < 16;
    const int      j    = lane & 15;         // column index == batch row within tile
    const unsigned rb   = (unsigned)((blockIdx.x * 16 + j) * T);  // fits in 32 bits

    // K pattern of the f32 2-VGPR A/B chunks: v0 -> K=(lo?0:2), v1 -> K=(lo?1:3)
    const int f0 = lo ? 0 : 2;
    const int f1 = lo ? 1 : 3;

    // Per-lane window base offsets (unsigned -> saddr+voffset scale_offset form)
    const unsigned uw0 = rb + (unsigned)f0;
    const unsigned uw1 = rb + (unsigned)f1;

    // A' chunks (constant): A'[i][k] = Wout[i&7][k] (rows duplicated into 8..15)
    const int i8 = lane & 7;
    v2f A0 = { Wout[i8 * 16 + 0 + f0],  Wout[i8 * 16 + 0 + f1] };
    v2f A1 = { Wout[i8 * 16 + 4 + f0],  Wout[i8 * 16 + 4 + f1] };
    v2f A2 = { Wout[i8 * 16 + 8 + f0],  Wout[i8 * 16 + 8 + f1] };
    v2f A3 = { Wout[i8 * 16 + 12 + f0], Wout[i8 * 16 + 12 + f1] };

    // Bias: with duplicated rows, cb[v] = bout[v] uniformly across lanes.
    const v8f cb = { bout[0], bout[1], bout[2], bout[3],
                     bout[4], bout[5], bout[6], bout[7] };

    // y dot coefficients (uniform -> SGPRs)
    const float w10 = w1[0], w11 = w1[1], w12 = w1[2], w13 = w1[3];
    const float w14 = w1[4], w15 = w1[5], w16 = w1[6], w17 = w1[7];
    const float b1v = b1p[0];

    // first L output columns zero, branchless: lo lanes cols 0-3, hi lanes 4-7
    {
        const unsigned zo = rb + (lo ? 0u : 4u);
#pragma unroll
        for (unsigned c = 0; c < 4; ++c) out[zo + c] = 0.f;
    }

    // o1_0 = 0
    float a0 = 0.f, a1 = 0.f, a2 = 0.f, a3 = 0.f,
          a4 = 0.f, a5 = 0.f, a6 = 0.f, a7 = 0.f;

    // Prologue: cpre = bias + ctx(0)^T x Wout-part
    v8f cpre;
    {
        v2f b0 = { in[uw0],     in[uw1] };
        v2f b1 = { in[uw0 + 4], in[uw1 + 4] };
        v8f p1 = WMMA_F32(A0, b0, cb);
        cpre   = WMMA_F32(A1, b1, p1);
    }

    // paired y store: lo lanes write column L+t, hi lanes column L+t+1 (same row)
    const unsigned yso = rb + (unsigned)L + (lo ? 0u : 1u);

    unsigned ut = 0;
    for (int t = 0; t + 1 < S; t += 2) {
        float ya, yb;
        STEP(ya, ut + 1u, true);    // step t,   prefetch/compute cpre for t+1
        STEP(yb, ut + 2u, true);    // step t+1, prefetch/compute cpre for t+2
        out[yso + ut] = lo ? ya : yb;   // all 32 lanes, unpredicated
        ut += 2;
    }

    if (S & 1) {                    // final step (t = S-1), no next cpre needed
        float yl;
        STEP(yl, 0u, false);
        if (lo) out[rb + (unsigned)(L + S - 1)] = yl;
    }
}

extern "C" void kernel_launch(void* const* d_in, const int* in_sizes, int n_in,
                              void* d_out, int out_size, void* d_ws, size_t ws_size,
                              hipStream_t stream) {
    const float* in   = (const float*)d_in[0];
    const float* Wout = (const float*)d_in[1];
    const float* bout = (const float*)d_in[2];
    const float* w1   = (const float*)d_in[3];
    const float* b1   = (const float*)d_in[4];
    float* out = (float*)d_out;

    const int N = 8192;                    // per reference setup
    const int T = in_sizes[0] / N;         // 999

    // 16 rows per wave32, 512 single-wave workgroups; register-resident
    // recurrence: cndmask -> 2 WMMA -> tanh per step, no LDS, no shuffles.
    att_rnn_wmma<<<dim3(N / 16), dim3(32), 0, stream>>>(in, Wout, bout, w1, b1, out, T);
}